// ST_GAT_50216757625084
// MI455X (gfx1250) — compile-verified
//
#include <hip/hip_runtime.h>
#include <math.h>

#define HEADS 8
#define CPH   9          // out channels per head
#define FOUT  72         // HEADS * CPH
#define FIN   12
#define NEG_SLOPE 0.2f

typedef float v2f __attribute__((ext_vector_type(2)));
typedef float v8f __attribute__((ext_vector_type(8)));

// ---------------------------------------------------------------------------
// Kernel 1: h[N,72] = x[N,12] @ W[12,72] using V_WMMA_F32_16X16X4_F32.
// One wave32 per 16-node tile; K=12 -> 3 k-steps of 4; 5 column tiles of 16.
// A layout (16x4 f32): lanes 0-15 hold K={0,1} in v0/v1, lanes 16-31 K={2,3}.
// B layout (4x16 f32): mirrored; column = lane%16.
// C layout (16x16 f32, 8 VGPRs): row = r + 8*(lane>=16), col = lane%16.
// Stores: full-tile fast path (wave-uniform) -> one base pointer per tile,
// 8 stores at immediate offsets r*288B; guards only on the generic path.
// ---------------------------------------------------------------------------
__global__ __launch_bounds__(256) void gat_gemm_wmma(
    const float* __restrict__ x, const float* __restrict__ W,
    float* __restrict__ h, int N, int numTiles)
{
  const int wave = threadIdx.x >> 5;
  const int lane = threadIdx.x & 31;
  const int tile = blockIdx.x * 8 + wave;
  if (tile >= numTiles) return;               // wave-uniform exit, EXEC stays full

  const int rowInTile = lane & 15;
  const int g         = lane >> 4;            // half-lane group (K offset 0 or 2)
  const int tileBase  = tile * 16;
  const int node      = tileBase + rowInTile;
  const bool fullTile = (tileBase + 16) <= N; // wave-uniform (true when N%16==0)
  const bool rowOK    = node < N;
  const int nodeC     = rowOK ? node : (N - 1);

  // Preload the A fragments for all 3 K-steps (predicated, no divergence).
  v2f a[3];
#pragma unroll
  for (int s = 0; s < 3; ++s) {
    const int k0 = 4 * s + 2 * g;
    a[s].x = rowOK ? x[nodeC * FIN + k0 + 0] : 0.0f;
    a[s].y = rowOK ? x[nodeC * FIN + k0 + 1] : 0.0f;
  }

#pragma unroll
  for (int ct = 0; ct < 5; ++ct) {            // output columns [16*ct, 16*ct+16)
    const int col    = 16 * ct + rowInTile;
    const bool colOK = col < FOUT;            // constant-true for ct<4 after unroll
    const int colC   = colOK ? col : (FOUT - 1);
    v8f acc = {};
#pragma unroll
    for (int s = 0; s < 3; ++s) {
      const int k0 = 4 * s + 2 * g;
      v2f b;
      b.x = colOK ? W[(k0 + 0) * FOUT + colC] : 0.0f;
      b.y = colOK ? W[(k0 + 1) * FOUT + colC] : 0.0f;
      acc = __builtin_amdgcn_wmma_f32_16x16x4_f32(
          /*neg_a=*/false, a[s], /*neg_b=*/false, b,
          /*c_mod=*/(short)0, acc, /*reuse_a=*/false, /*reuse_b=*/false);
    }
    if (colOK) {                              // single exec-mask set per tile
      float* hp = h + (size_t)(tileBase + 8 * g) * FOUT + col;
      if (fullTile) {
#pragma unroll
        for (int r = 0; r < 8; ++r) hp[(size_t)r * FOUT] = acc[r];  // imm offsets
      } else {
#pragma unroll
        for (int r = 0; r < 8; ++r) {
          if (tileBase + 8 * g + r < N) hp[(size_t)r * FOUT] = acc[r];
        }
      }
    }
  }
}

// ---------------------------------------------------------------------------
// Kernel 2: per-node attention halves + init of max/denom/accumulator.
// ---------------------------------------------------------------------------
__global__ __launch_bounds__(256) void gat_node_init(
    const float* __restrict__ h,
    const float* __restrict__ a_src, const float* __restrict__ a_dst,
    float* __restrict__ al_src, float* __restrict__ al_dst,
    float* __restrict__ mmax, float* __restrict__ denom,
    float* __restrict__ out_acc, int N)
{
  const int n = blockIdx.x * blockDim.x + threadIdx.x;
  if (n >= N) return;
  const float* hn = h + (size_t)n * FOUT;
#pragma unroll
  for (int hd = 0; hd < HEADS; ++hd) {
    float s0 = 0.0f, s1 = 0.0f;
#pragma unroll
    for (int c = 0; c < CPH; ++c) {
      const float v = hn[hd * CPH + c];
      s0 = fmaf(v, a_src[hd * CPH + c], s0);
      s1 = fmaf(v, a_dst[hd * CPH + c], s1);
    }
    al_src[n * HEADS + hd] = s0;
    al_dst[n * HEADS + hd] = s1;
    mmax [n * HEADS + hd] = -INFINITY;
    denom[n * HEADS + hd] = 0.0f;
  }
#pragma unroll
  for (int c = 0; c < FOUT; ++c) out_acc[(size_t)n * FOUT + c] = 0.0f;
}

// Exact float atomic-max via sign-aware integer atomics (valid with -inf init).
__device__ __forceinline__ void atomicMaxF(float* addr, float val) {
  if (!(val < 0.0f)) atomicMax((int*)addr, __float_as_int(val));
  else               atomicMin((unsigned int*)addr, (unsigned int)__float_as_int(val));
}

__device__ __forceinline__ float leaky(float v) {
  return v > 0.0f ? v : NEG_SLOPE * v;
}

// ---------------------------------------------------------------------------
// Kernel 3: segment-max of edge logits over destination nodes.
// ---------------------------------------------------------------------------
__global__ __launch_bounds__(256) void gat_edge_max(
    const int* __restrict__ srcIdx, const int* __restrict__ dstIdx,
    const float* __restrict__ al_src, const float* __restrict__ al_dst,
    float* __restrict__ mmax, int E, int N)
{
  const int e = blockIdx.x * blockDim.x + threadIdx.x;
  const int Etot = E + N;
  if (e >= Etot) return;
  const int s = (e < E) ? srcIdx[e] : (e - E);   // self-loops appended
  const int d = (e < E) ? dstIdx[e] : (e - E);
#pragma unroll
  for (int hd = 0; hd < HEADS; ++hd) {
    const float v = leaky(al_src[s * HEADS + hd] + al_dst[d * HEADS + hd]);
    atomicMaxF(&mmax[d * HEADS + hd], v);
  }
}

// ---------------------------------------------------------------------------
// Kernel 4: segment-sum of exp(logit - max).
// ---------------------------------------------------------------------------
__global__ __launch_bounds__(256) void gat_edge_expsum(
    const int* __restrict__ srcIdx, const int* __restrict__ dstIdx,
    const float* __restrict__ al_src, const float* __restrict__ al_dst,
    const float* __restrict__ mmax, float* __restrict__ denom, int E, int N)
{
  const int e = blockIdx.x * blockDim.x + threadIdx.x;
  const int Etot = E + N;
  if (e >= Etot) return;
  const int s = (e < E) ? srcIdx[e] : (e - E);
  const int d = (e < E) ? dstIdx[e] : (e - E);
#pragma unroll
  for (int hd = 0; hd < HEADS; ++hd) {
    const float v  = leaky(al_src[s * HEADS + hd] + al_dst[d * HEADS + hd]);
    const float ex = __expf(v - mmax[d * HEADS + hd]);
    atomicAdd(&denom[d * HEADS + hd], ex);
  }
}

// ---------------------------------------------------------------------------
// Kernel 5: alpha-weighted scatter aggregate; one thread per (edge, head).
// ---------------------------------------------------------------------------
__global__ __launch_bounds__(256) void gat_edge_aggregate(
    const int* __restrict__ srcIdx, const int* __restrict__ dstIdx,
    const float* __restrict__ al_src, const float* __restrict__ al_dst,
    const float* __restrict__ mmax, const float* __restrict__ denom,
    const float* __restrict__ h, float* __restrict__ out_acc, int E, int N)
{
  const long long t = (long long)blockIdx.x * blockDim.x + threadIdx.x;
  const long long Ttot = (long long)(E + N) * HEADS;
  if (t >= Ttot) return;
  const int e  = (int)(t >> 3);
  const int hd = (int)(t & 7);
  const int s = (e < E) ? srcIdx[e] : (e - E);
  const int d = (e < E) ? dstIdx[e] : (e - E);
  const float v     = leaky(al_src[s * HEADS + hd] + al_dst[d * HEADS + hd]);
  const float alpha = __expf(v - mmax[d * HEADS + hd]) /
                      (denom[d * HEADS + hd] + 1e-16f);
  const float* hs = h       + (size_t)s * FOUT + hd * CPH;
  float*       oa = out_acc + (size_t)d * FOUT + hd * CPH;
#pragma unroll
  for (int c = 0; c < CPH; ++c) atomicAdd(&oa[c], alpha * hs[c]);
}

// ---------------------------------------------------------------------------
// Kernel 6: head-mean + bias + log_softmax over the 9 output channels.
// ---------------------------------------------------------------------------
__global__ __launch_bounds__(256) void gat_finalize(
    const float* __restrict__ out_acc, const float* __restrict__ bias,
    float* __restrict__ out, int N)
{
  const int n = blockIdx.x * blockDim.x + threadIdx.x;
  if (n >= N) return;
  float o[CPH];
#pragma unroll
  for (int c = 0; c < CPH; ++c) {
    float sum = 0.0f;
#pragma unroll
    for (int hd = 0; hd < HEADS; ++hd) sum += out_acc[(size_t)n * FOUT + hd * CPH + c];
    o[c] = sum * (1.0f / HEADS) + bias[c];
  }
  float mx = o[0];
#pragma unroll
  for (int c = 1; c < CPH; ++c) mx = fmaxf(mx, o[c]);
  float se = 0.0f;
#pragma unroll
  for (int c = 0; c < CPH; ++c) se += __expf(o[c] - mx);
  const float lse = mx + __logf(se);
#pragma unroll
  for (int c = 0; c < CPH; ++c) out[(size_t)n * CPH + c] = o[c] - lse;
}

// ---------------------------------------------------------------------------
extern "C" void kernel_launch(void* const* d_in, const int* in_sizes, int n_in,
                              void* d_out, int out_size, void* d_ws, size_t ws_size,
                              hipStream_t stream)
{
  const float* x     = (const float*)d_in[0];
  const int*   ei    = (const int*)  d_in[1];
  const float* W     = (const float*)d_in[2];
  const float* a_src = (const float*)d_in[3];
  const float* a_dst = (const float*)d_in[4];
  const float* bias  = (const float*)d_in[5];

  const int N = in_sizes[0] / FIN;
  const int E = in_sizes[1] / 2;
  const int* srcIdx = ei;
  const int* dstIdx = ei + E;

  // Workspace layout (floats): h | al_src | al_dst | mmax | denom | out_acc
  float* ws = (float*)d_ws;
  size_t off = 0;
  float* h       = ws + off; off += (size_t)N * FOUT;
  float* al_src  = ws + off; off += (size_t)N * HEADS;
  float* al_dst  = ws + off; off += (size_t)N * HEADS;
  float* mmax    = ws + off; off += (size_t)N * HEADS;
  float* denom   = ws + off; off += (size_t)N * HEADS;
  float* out_acc = ws + off; off += (size_t)N * FOUT;

  const int numTiles = (N + 15) / 16;
  gat_gemm_wmma<<<(numTiles + 7) / 8, 256, 0, stream>>>(x, W, h, N, numTiles);
  gat_node_init<<<(N + 255) / 256, 256, 0, stream>>>(h, a_src, a_dst,
      al_src, al_dst, mmax, denom, out_acc, N);

  const int Etot = E + N;
  gat_edge_max   <<<(Etot + 255) / 256, 256, 0, stream>>>(srcIdx, dstIdx,
      al_src, al_dst, mmax, E, N);
  gat_edge_expsum<<<(Etot + 255) / 256, 256, 0, stream>>>(srcIdx, dstIdx,
      al_src, al_dst, mmax, denom, E, N);

  const long long T = (long long)Etot * HEADS;
  gat_edge_aggregate<<<(int)((T + 255) / 256), 256, 0, stream>>>(srcIdx, dstIdx,
      al_src, al_dst, mmax, denom, h, out_acc, E, N);

  gat_finalize<<<(N + 255) / 256, 256, 0, stream>>>(out_acc, bias,
      (float*)d_out, N);
}